// SigmoidTopKRouter_76536317215267
// MI455X (gfx1250) — compile-verified
//
#include <hip/hip_runtime.h>

typedef __attribute__((ext_vector_type(2))) float v2f;
typedef __attribute__((ext_vector_type(8))) float v8f;

#define NUM_TOKENS   16384
#define DIM          4096
#define NUM_EXPERTS  64
#define TOPK         8
#define TOK_PER_WAVE 16
#define WAVES_PER_BLK 8
#define TOK_PER_BLK  128
#define LDS_STRIDE   65   // 64 + 1 pad -> conflict-free LDS rows

// ---------------------------------------------------------------------------
// Kernel 1: repack gate_w (64 x 4096, row-major) into a K-pair interleaved
// transposed layout so WMMA B-fragments are contiguous float2 per lane:
//   wsB[(k>>1)*128 + e*2 + (k&1)] = gate_w[e][k]
// B fragment for k-step k0, lane L: rows K = 2*(L>>4) + {0,1}, col e,
// i.e. one float2 at wsB + ((k0>>1)+(L>>4))*128 + e*2. Lanes 0..15 of an
// expert tile then read one contiguous 128B segment. One-time 1 MB shuffle.
// ---------------------------------------------------------------------------
__global__ __launch_bounds__(256) void pack_gate_kernel(
    const float* __restrict__ gate_w, float* __restrict__ wsB) {
  int idx = blockIdx.x * blockDim.x + threadIdx.x;  // 0 .. 64*4096-1
  int e = idx >> 12;        // expert  (coalesced read along k)
  int k = idx & (DIM - 1);  // dim
  wsB[(size_t)(k >> 1) * 128 + e * 2 + (k & 1)] = gate_w[(size_t)e * DIM + k];
}

// ---------------------------------------------------------------------------
// Kernel 2: logits via V_WMMA_F32_16X16X4_F32, sigmoid+bias -> LDS,
// per-token top-8 insertion select, normalize, emit weights + indices.
// ---------------------------------------------------------------------------
__global__ __launch_bounds__(256) void router_kernel(
    const float* __restrict__ x,
    const float* __restrict__ wsB,
    const float* __restrict__ bias,
    float* __restrict__ out_w,
    float* __restrict__ out_i) {
  __shared__ float s_scores[TOK_PER_BLK * LDS_STRIDE];

  const int tid   = threadIdx.x;
  const int wave  = tid >> 5;
  const int lane  = tid & 31;
  const int lhalf = lane >> 4;   // 0: K=0,1 / C rows 0-7   1: K=2,3 / C rows 8-15
  const int lmod  = lane & 15;

  const int tok_base = blockIdx.x * TOK_PER_BLK + wave * TOK_PER_WAVE;

  // A: lane holds x[tok_base + lmod][k0 + 2*lhalf + {0,1}] as float2
  const float* ax = x + (size_t)(tok_base + lmod) * DIM + (lhalf << 1);
  // B: lane holds wsB[((k0>>1)+lhalf)*128 + (t*16+lmod)*2 + {0,1}]
  const float* bx = wsB + (size_t)lhalf * 128 + lmod * 2;

  v8f acc[4];
  acc[0] = (v8f){}; acc[1] = (v8f){}; acc[2] = (v8f){}; acc[3] = (v8f){};

  #pragma unroll 4
  for (int k0 = 0; k0 < DIM; k0 += 4) {
    v2f a  = *(const v2f*)(ax); ax += 4;
    v2f b0 = *(const v2f*)(bx);
    v2f b1 = *(const v2f*)(bx + 32);
    v2f b2 = *(const v2f*)(bx + 64);
    v2f b3 = *(const v2f*)(bx + 96);
    bx += 256;  // k-pair advances by 2 per k-step of 4
    acc[0] = __builtin_amdgcn_wmma_f32_16x16x4_f32(false, a, false, b0,
                                                   (short)0, acc[0], false, false);
    acc[1] = __builtin_amdgcn_wmma_f32_16x16x4_f32(false, a, false, b1,
                                                   (short)0, acc[1], false, false);
    acc[2] = __builtin_amdgcn_wmma_f32_16x16x4_f32(false, a, false, b2,
                                                   (short)0, acc[2], false, false);
    acc[3] = __builtin_amdgcn_wmma_f32_16x16x4_f32(false, a, false, b3,
                                                   (short)0, acc[3], false, false);
  }

  // Per-lane bias for its expert column in each 16-expert tile.
  float bv[4];
  #pragma unroll
  for (int t = 0; t < 4; ++t) bv[t] = bias[t * 16 + lmod];

  // C layout: VGPR r, lanes 0-15 -> row M=r; lanes 16-31 -> row M=8+r; N=lmod.
  const int mrow = wave * TOK_PER_WAVE + lhalf * 8;
  #pragma unroll
  for (int t = 0; t < 4; ++t) {
    #pragma unroll
    for (int r = 0; r < 8; ++r) {
      float z = acc[t][r] + bv[t];
      float s = 1.0f / (1.0f + __expf(-z));        // sigmoid
      s_scores[(mrow + r) * LDS_STRIDE + t * 16 + lmod] = s;
    }
  }

  __syncthreads();

  // One thread per token: stable top-8 insertion over 64 experts.
  if (tid < TOK_PER_BLK) {
    const float* row = &s_scores[tid * LDS_STRIDE];
    float wv[TOPK]; int wi[TOPK];
    #pragma unroll
    for (int j = 0; j < TOPK; ++j) { wv[j] = -3.0e38f; wi[j] = 0; }

    for (int e = 0; e < NUM_EXPERTS; ++e) {
      float cv = row[e];
      int   ci = e;
      #pragma unroll
      for (int j = 0; j < TOPK; ++j) {
        // strict '>' keeps earlier (lower) index on ties, matching lax.top_k
        if (cv > wv[j]) {
          float tv = wv[j]; int ti = wi[j];
          wv[j] = cv; wi[j] = ci;
          cv = tv;    ci = ti;
        }
      }
    }

    float sum = 0.0f;
    #pragma unroll
    for (int j = 0; j < TOPK; ++j) sum += wv[j];
    float inv = 1.0f / sum;

    const int g = blockIdx.x * TOK_PER_BLK + tid;
    #pragma unroll
    for (int j = 0; j < TOPK; ++j) {
      out_w[g * TOPK + j] = wv[j] * inv;
      out_i[g * TOPK + j] = (float)wi[j];  // tuple concat promotes idx to f32
    }
  }
}

extern "C" void kernel_launch(void* const* d_in, const int* in_sizes, int n_in,
                              void* d_out, int out_size, void* d_ws, size_t ws_size,
                              hipStream_t stream) {
  const float* x      = (const float*)d_in[0];
  const float* gate_w = (const float*)d_in[1];
  const float* bias   = (const float*)d_in[2];
  // d_in[3] = top_k scalar (== 8, compile-time constant here)

  float* wsB   = (float*)d_ws;                       // 64*4096 floats = 1 MB
  float* out_w = (float*)d_out;                      // (16384, 8) weights
  float* out_i = out_w + (size_t)NUM_TOKENS * TOPK;  // (16384, 8) indices

  pack_gate_kernel<<<(NUM_EXPERTS * DIM) / 256, 256, 0, stream>>>(gate_w, wsB);
  router_kernel<<<NUM_TOKENS / TOK_PER_BLK, 256, 0, stream>>>(
      x, wsB, bias, out_w, out_i);
}